// GraphGenerate_197568496061
// MI455X (gfx1250) — compile-verified
//
#include <hip/hip_runtime.h>

// ---------------------------------------------------------------------------
// Shapes (fixed by the reference):
//   B=4, T=512, C=256, 2C=512, HEADS=4
//   x_new : [B*T=2048][512]   (d_out part 0, 1,048,576 f32)
//   attn  : [B][T][T][4]      (d_out part 1, 4,194,304 f32)
// Workspace:
//   subj  [2048][256] f32 @ ws+0
//   obj   [2048][256] f32 @ ws+524288
//   psum  [NCHUNK=8][2048][4] f32 @ ws+1048576  (partial softmax denominators)
// ---------------------------------------------------------------------------

typedef __attribute__((ext_vector_type(2))) float v2f;
typedef __attribute__((ext_vector_type(8))) float v8f;

#define NB     4
#define NT     512
#define NC     256
#define NK     512          // 2*C
#define NH     4
#define NCHUNK 8            // j-range split for occupancy (4 j-tiles each)
#define LSTR   260          // LDS row stride (260 % 64 == 4 -> conflict-free)

// --------------------------- CDNA5 async helpers ---------------------------
// Async global->LDS copy (VGLOBAL GLOBAL_LOAD_ASYNC_TO_LDS_B128, ASYNCcnt).
// LDS destination address = low 32 bits of the generic pointer.
__device__ __forceinline__ void async_copy_b128(const float* gsrc, float* lds_dst)
{
    unsigned lds = (unsigned)(unsigned long long)lds_dst;
    asm volatile("global_load_async_to_lds_b128 %0, %1, off"
                 :: "v"(lds), "v"(gsrc) : "memory");
}
template <int N>
__device__ __forceinline__ void wait_asynccnt()
{
    asm volatile("s_wait_asynccnt %0" :: "i"(N) : "memory");
}

// ---------------------------------------------------------------------------
// Kernel A: channel max-pool (window 5, same pad) + concat -> x_new in d_out.
// ---------------------------------------------------------------------------
__global__ __launch_bounds__(256) void pool_concat_kernel(
    const float* __restrict__ x, float* __restrict__ xnew)
{
    __shared__ float sx[NC];
    const int bt = blockIdx.x;            // 0..2047
    const int c  = threadIdx.x;           // 0..255
    const float v = x[bt * NC + c];
    sx[c] = v;
    __syncthreads();
    float m = v;
    #pragma unroll
    for (int off = -2; off <= 2; ++off) {
        int cc = c + off;
        if (cc >= 0 && cc < NC) m = fmaxf(m, sx[cc]);
    }
    xnew[bt * NK + c]      = m;   // pooled half
    xnew[bt * NK + NC + c] = v;   // raw half
}

// ---------------------------------------------------------------------------
// Kernel B: subj = x_new @ W_subj + b_subj ; obj = x_new @ W_obj + b_obj
// Full-precision f32 WMMA (V_WMMA_F32_16X16X4_F32), one 16x16 tile per wave.
// ---------------------------------------------------------------------------
__global__ __launch_bounds__(256) void gemm_subj_obj_kernel(
    const float* __restrict__ xnew,
    const float* __restrict__ Wsubj, const float* __restrict__ bsubj,
    const float* __restrict__ Wobj,  const float* __restrict__ bobj,
    float* __restrict__ subj, float* __restrict__ obj)
{
    const int lane   = threadIdx.x & 31;
    const int waveId = blockIdx.x * 8 + (threadIdx.x >> 5);   // 0..4095
    const int which  = waveId >> 11;                          // 0=subj, 1=obj
    const int rem    = waveId & 2047;
    const int mt     = rem >> 4;                              // 0..127
    const int nt     = rem & 15;                              // 0..15

    const float* __restrict__ W    = which ? Wobj : Wsubj;
    const float* __restrict__ bias = which ? bobj : bsubj;
    float* __restrict__ out        = which ? obj  : subj;

    const int hi  = lane >> 4;           // half-wave select (K pairs)
    const int l15 = lane & 15;
    const int row = mt * 16 + l15;       // A row for this lane
    const int col = nt * 16 + l15;       // B/D col for this lane

    const float* __restrict__ aptr = xnew + row * NK;
    v8f acc = {};
    for (int k = 0; k < NK; k += 4) {
        const float2 av = *(const float2*)(aptr + k + 2 * hi);   // K, K+1
        v2f a; a.x = av.x; a.y = av.y;
        v2f b;
        b.x = W[(k + 2 * hi)     * NC + col];
        b.y = W[(k + 2 * hi + 1) * NC + col];
        acc = __builtin_amdgcn_wmma_f32_16x16x4_f32(
            false, a, false, b, (short)0, acc, false, false);
    }
    const float bv    = bias[col];
    const int   mbase = mt * 16 + 8 * hi;
    #pragma unroll
    for (int r = 0; r < 8; ++r)
        out[(mbase + r) * NC + col] = acc[r] + bv;
}

// ---------------------------------------------------------------------------
// Kernel C: hot loop. Block tile = 32 i x 64 j; thread (il,jl) owns rows
// (il, il+16) x col jl. Per channel: 3x ds_load_b128 feed 8 FMAs; W_t is
// lane-uniform -> SMEM (s_load) into SGPR operands, no DS traffic.
// All 4 subj j-tiles are issued up-front with global_load_async_to_lds_b128
// and drained with staged s_wait_asynccnt (in-order completion) so tile n+1..3
// loads overlap tile-n compute.
// Grid: b(4) x i-tile(16) x j-chunk(8) = 512 blocks, 256 threads.
// ---------------------------------------------------------------------------
__global__ __launch_bounds__(256) void pairwise_softmax_kernel(
    const float* __restrict__ subj, const float* __restrict__ obj,
    const float* __restrict__ Wt,   const float* __restrict__ bt,
    float* __restrict__ attn, float* __restrict__ psum)
{
    __shared__ float o_s[32 * LSTR];      // 32 obj rows
    __shared__ float s_s[64 * LSTR];      // 4 subj tiles x 16 rows
    __shared__ float red[32 * 16 * NH];   // partial-sum reduction

    const int tid = threadIdx.x;
    const int jc  = blockIdx.x & 7;              // j-chunk (4 j-tiles)
    const int it  = (blockIdx.x >> 3) & 15;      // i-tile (32 rows)
    const int b   = blockIdx.x >> 7;             // batch
    const int il  = tid >> 4;                    // 0..15
    const int jl  = tid & 15;                    // 0..15
    const int i0  = it * 32 + il;                // first owned i row
    const int i1  = i0 + 16;                     // second owned i row

    const float4* __restrict__ Wt4 = (const float4*)Wt;   // uniform -> SMEM
    const float bt0 = bt[0], bt1 = bt[1], bt2 = bt[2], bt3 = bt[3];

    // ---- issue ALL async copies up-front (8 obj + 4x4 subj per wave) ----
    #pragma unroll
    for (int q = 0; q < 8; ++q) {                 // obj: 32 rows x 256 ch
        int e = q * 1024 + tid * 4;
        int r = e >> 8, c = e & 255;
        async_copy_b128(obj + (b * NT + it * 32 + r) * NC + c,
                        &o_s[r * LSTR + c]);
    }
    #pragma unroll
    for (int n = 0; n < 4; ++n) {                 // subj tiles jc*4+n
        #pragma unroll
        for (int q = 0; q < 4; ++q) {
            int e = q * 1024 + tid * 4;
            int r = e >> 8, c = e & 255;
            async_copy_b128(subj + (b * NT + (jc * 4 + n) * 16 + r) * NC + c,
                            &s_s[(n * 16 + r) * LSTR + c]);
        }
    }

    float acc[8] = {0.f, 0.f, 0.f, 0.f, 0.f, 0.f, 0.f, 0.f};

    #pragma unroll
    for (int n = 0; n < 4; ++n) {
        // drain: obj + subj tiles 0..n complete (in-order), rest in flight
        if      (n == 0) wait_asynccnt<12>();
        else if (n == 1) wait_asynccnt<8>();
        else if (n == 2) wait_asynccnt<4>();
        else             wait_asynccnt<0>();
        __syncthreads();

        const float* __restrict__ srow = &s_s[(n * 16 + jl) * LSTR];
        const float* __restrict__ or0  = &o_s[il * LSTR];
        const float* __restrict__ or1  = &o_s[(il + 16) * LSTR];

        float r00=0.f,r01=0.f,r02=0.f,r03=0.f;    // row i0, heads 0..3
        float r10=0.f,r11=0.f,r12=0.f,r13=0.f;    // row i1, heads 0..3
        #pragma unroll 2
        for (int c4 = 0; c4 < NC / 4; ++c4) {
            const float4 s4 = *(const float4*)(srow + 4 * c4);
            const float4 a4 = *(const float4*)(or0  + 4 * c4);
            const float4 b4 = *(const float4*)(or1  + 4 * c4);
            const float4 w0 = Wt4[4 * c4 + 0];
            const float4 w1 = Wt4[4 * c4 + 1];
            const float4 w2 = Wt4[4 * c4 + 2];
            const float4 w3 = Wt4[4 * c4 + 3];
            float d;
            d = fabsf(s4.x - a4.x);
            r00 = fmaf(d, w0.x, r00); r01 = fmaf(d, w0.y, r01);
            r02 = fmaf(d, w0.z, r02); r03 = fmaf(d, w0.w, r03);
            d = fabsf(s4.x - b4.x);
            r10 = fmaf(d, w0.x, r10); r11 = fmaf(d, w0.y, r11);
            r12 = fmaf(d, w0.z, r12); r13 = fmaf(d, w0.w, r13);
            d = fabsf(s4.y - a4.y);
            r00 = fmaf(d, w1.x, r00); r01 = fmaf(d, w1.y, r01);
            r02 = fmaf(d, w1.z, r02); r03 = fmaf(d, w1.w, r03);
            d = fabsf(s4.y - b4.y);
            r10 = fmaf(d, w1.x, r10); r11 = fmaf(d, w1.y, r11);
            r12 = fmaf(d, w1.z, r12); r13 = fmaf(d, w1.w, r13);
            d = fabsf(s4.z - a4.z);
            r00 = fmaf(d, w2.x, r00); r01 = fmaf(d, w2.y, r01);
            r02 = fmaf(d, w2.z, r02); r03 = fmaf(d, w2.w, r03);
            d = fabsf(s4.z - b4.z);
            r10 = fmaf(d, w2.x, r10); r11 = fmaf(d, w2.y, r11);
            r12 = fmaf(d, w2.z, r12); r13 = fmaf(d, w2.w, r13);
            d = fabsf(s4.w - a4.w);
            r00 = fmaf(d, w3.x, r00); r01 = fmaf(d, w3.y, r01);
            r02 = fmaf(d, w3.z, r02); r03 = fmaf(d, w3.w, r03);
            d = fabsf(s4.w - b4.w);
            r10 = fmaf(d, w3.x, r10); r11 = fmaf(d, w3.y, r11);
            r12 = fmaf(d, w3.z, r12); r13 = fmaf(d, w3.w, r13);
        }
        const float e00 = __expf(fmaxf(r00 + bt0, 0.f));
        const float e01 = __expf(fmaxf(r01 + bt1, 0.f));
        const float e02 = __expf(fmaxf(r02 + bt2, 0.f));
        const float e03 = __expf(fmaxf(r03 + bt3, 0.f));
        const float e10 = __expf(fmaxf(r10 + bt0, 0.f));
        const float e11 = __expf(fmaxf(r11 + bt1, 0.f));
        const float e12 = __expf(fmaxf(r12 + bt2, 0.f));
        const float e13 = __expf(fmaxf(r13 + bt3, 0.f));
        const int j_g = (jc * 4 + n) * 16 + jl;
        float4 ev0; ev0.x = e00; ev0.y = e01; ev0.z = e02; ev0.w = e03;
        float4 ev1; ev1.x = e10; ev1.y = e11; ev1.z = e12; ev1.w = e13;
        *(float4*)(attn + ((b * NT + i0) * NT + j_g) * NH) = ev0;
        *(float4*)(attn + ((b * NT + i1) * NT + j_g) * NH) = ev1;
        acc[0] += e00; acc[1] += e01; acc[2] += e02; acc[3] += e03;
        acc[4] += e10; acc[5] += e11; acc[6] += e12; acc[7] += e13;
    }

    // ---- reduce partial softmax denominators over the 16 j_locals ----
    #pragma unroll
    for (int h = 0; h < NH; ++h) {
        red[(il * 16 + jl) * NH + h]        = acc[h];
        red[((il + 16) * 16 + jl) * NH + h] = acc[4 + h];
    }
    __syncthreads();
    if (tid < 128) {
        const int i = tid >> 2;        // 0..31
        const int h = tid & 3;
        float s = 0.f;
        #pragma unroll
        for (int k = 0; k < 16; ++k) s += red[(i * 16 + k) * NH + h];
        psum[((jc * (NB * NT)) + b * NT + it * 32 + i) * NH + h] = s;
    }
}

// ---------------------------------------------------------------------------
// Kernel D: attn = mm ? 0 : exp/sum. Sum over the NCHUNK partials. Pure BW.
// ---------------------------------------------------------------------------
__global__ __launch_bounds__(256) void normalize_mask_kernel(
    float* __restrict__ attn, const float* __restrict__ psum,
    const unsigned char* __restrict__ mask)
{
    const int idx = blockIdx.x * 256 + threadIdx.x;   // 0..1048575
    const int b = idx >> 18;
    const int i = (idx >> 9) & (NT - 1);
    const int j = idx & (NT - 1);

    float4 s; s.x = 0.f; s.y = 0.f; s.z = 0.f; s.w = 0.f;
    #pragma unroll
    for (int c = 0; c < NCHUNK; ++c) {
        const float4 p = *(const float4*)(psum + ((c * (NB * NT)) + b * NT + i) * NH);
        s.x += p.x; s.y += p.y; s.z += p.z; s.w += p.w;
    }
    const bool mm = (mask[b * NT + i] != 0) && (mask[b * NT + j] != 0);
    float4 e = *(float4*)(attn + idx * NH);
    float4 r;
    r.x = mm ? 0.f : e.x / s.x;
    r.y = mm ? 0.f : e.y / s.y;
    r.z = mm ? 0.f : e.z / s.z;
    r.w = mm ? 0.f : e.w / s.w;
    *(float4*)(attn + idx * NH) = r;
}

// ---------------------------------------------------------------------------
extern "C" void kernel_launch(void* const* d_in, const int* in_sizes, int n_in,
                              void* d_out, int out_size, void* d_ws, size_t ws_size,
                              hipStream_t stream)
{
    const float* x     = (const float*)d_in[0];
    const float* Wsubj = (const float*)d_in[1];
    const float* bsubj = (const float*)d_in[2];
    const float* Wobj  = (const float*)d_in[3];
    const float* bobj  = (const float*)d_in[4];
    const float* Wt    = (const float*)d_in[5];
    const float* bt    = (const float*)d_in[6];
    const unsigned char* mask = (const unsigned char*)d_in[7]; // jnp bool = 1B

    float* out  = (float*)d_out;
    float* xnew = out;                         // 2048*512
    float* attn = out + (NB * NT) * NK;        // 2048*512*4

    float* ws   = (float*)d_ws;
    float* subj = ws;                          // 2048*256
    float* obj  = ws + (NB * NT) * NC;         // 2048*256
    float* psum = ws + 2 * (NB * NT) * NC;     // 8*2048*4

    pool_concat_kernel<<<NB * NT, 256, 0, stream>>>(x, xnew);
    gemm_subj_obj_kernel<<<512, 256, 0, stream>>>(xnew, Wsubj, bsubj,
                                                  Wobj, bobj, subj, obj);
    pairwise_softmax_kernel<<<NB * 16 * NCHUNK, 256, 0, stream>>>(
        subj, obj, Wt, bt, attn, psum);
    normalize_mask_kernel<<<(NB * NT * NT) / 256, 256, 0, stream>>>(
        attn, psum, mask);
}